// QuantizerDepthwiseBase_30253749633754
// MI455X (gfx1250) — compile-verified
//
#include <hip/hip_runtime.h>
#include <stdint.h>

// Binarized (±1) depthwise 3x3 conv, stride 1, SAME, NHWC.
//   x: (16,112,112,256) f32, k: (3,3,256,1) f32, out: (16,112,112,256) f32
// Memory-bound (AI ~1.1 FLOP/B): roofline = ~410 MB HBM traffic @ 23.3 TB/s
// -> ~17.6 us. CDNA5 path: the 3-row stencil window is staged into LDS by
// the Tensor Data Mover (one tensor_load_to_lds descriptor per workgroup,
// issued by wave 0, completion via S_WAIT_TENSORCNT), so each input byte is
// fetched once and the 9-tap reuse is served from LDS while compute waves
// spend zero issue slots on copies.

#define N_  16
#define H_  112
#define W_  112
#define C_  256
#define CG_ 32                       // channels per workgroup tile
#define NCG (C_ / CG_)               // 8 channel groups
#define ROW_ELEMS (W_ * CG_)         // 3584 floats per staged row slice
#define LDS_ELEMS (3 * ROW_ELEMS)    // 10752 floats = 42 KB

typedef unsigned int v4u __attribute__((ext_vector_type(4)));
typedef int          v8i __attribute__((ext_vector_type(8)));
typedef int          v4i __attribute__((ext_vector_type(4)));

#define BIG_DIM 0x40000000           // 2^30: OOB zero-fill never triggers

__global__ __launch_bounds__(256)
void bsign_dwconv3x3_kernel(const float* __restrict__ x,
                            const float* __restrict__ k,
                            float* __restrict__ out) {
    __shared__ float smem[LDS_ELEMS];

    const int bid = blockIdx.x;
    const int cg  = bid & (NCG - 1);           // channel group 0..7
    const int h   = (bid >> 3) % H_;           // output row
    const int n   = bid / (NCG * H_);          // image
    const int tid = threadIdx.x;

    // ---------------------------------------------------------------
    // TDM stage: one 3D tile = 32 contiguous floats (dim0, c) x 112
    // (dim1, w, stride 256 floats) x nrows (dim2, h, stride 112*256
    // floats) -> LDS row-major [r][w][c]. Rows clamped at image edges;
    // LDS base shifted so LDS row r always maps to image row h-1+r.
    // Issued by wave 0 only (TDM ignores EXEC; one DMA per workgroup).
    // ---------------------------------------------------------------
    if (tid < 32) {
        const int row0  = (h - 1 < 0) ? 0 : h - 1;
        const int row1  = (h + 1 > H_ - 1) ? H_ - 1 : h + 1;
        const int nrows = row1 - row0 + 1;                 // 2 or 3

        const uint64_t gaddr =
            (uint64_t)(uintptr_t)(x + ((size_t)(n * H_ + row0) * W_) * C_
                                    + (size_t)cg * CG_);
        const unsigned lds_base =
            (unsigned)(uintptr_t)&smem[(row0 - (h - 1)) * ROW_ELEMS];

        // ---- D# group 0: count=1, lds_addr, global_addr, type=2 ----
        v4u g0;
        g0.x = 1u;                                         // count=1, user mode
        g0.y = lds_base;                                   // LDS byte address
        g0.z = (unsigned)(gaddr & 0xFFFFFFFFull);          // global_addr[31:0]
        g0.w = (unsigned)((gaddr >> 32) & 0x1FFFFFFull)    // global_addr[56:32]
             | 0x80000000u;                                // type=2 ("image")

        // ---- D# group 1: sizes/strides ----
        v8i g1;
        g1[0] = (2 << 16);                                 // wg_mask=0, data_size=4B
        g1[1] = (int)((BIG_DIM & 0xFFFF) << 16);           // tensor_dim0[15:0]
        g1[2] = (int)(((unsigned)BIG_DIM >> 16)            // tensor_dim0[31:16]
              | ((BIG_DIM & 0xFFFF) << 16));               // tensor_dim1[15:0]
        g1[3] = (int)(((unsigned)BIG_DIM >> 16)            // tensor_dim1[31:16]
              | ((unsigned)CG_ << 16));                    // tile_dim0 = 32
        g1[4] = (int)((unsigned)W_ | ((unsigned)nrows << 16)); // tile_dim1=112, tile_dim2
        g1[5] = C_;                                        // tensor_dim0_stride = 256
        g1[6] = (int)(((unsigned)(W_ * C_) & 0xFFFF) << 16); // dim1_stride[15:0]
        g1[7] = (int)((unsigned)(W_ * C_) >> 16);          // dim1_stride[47:16]

        // ---- D# group 2: tensor_dim2 large; no dim3, no iterate ----
        v4i g2; g2[0] = BIG_DIM; g2[1] = 0; g2[2] = 0; g2[3] = 0;
        // ---- D# group 3: no dim3 stride / dim4 ----
        v4i g3; g3[0] = 0; g3[1] = 0; g3[2] = 0; g3[3] = 0;
        // ---- 5th operand of the clang-23 6-arg builtin (zero-filled,
        //      matches the probe-verified call shape) ----
        v8i g4; g4[0]=0; g4[1]=0; g4[2]=0; g4[3]=0; g4[4]=0; g4[5]=0; g4[6]=0; g4[7]=0;

        __builtin_amdgcn_tensor_load_to_lds(g0, g1, g2, g3, g4, /*cpol=*/0);
        __builtin_amdgcn_s_wait_tensorcnt(0);
    }
    __syncthreads();

    // ---------------------------------------------------------------
    // Per-lane kernel sign bits for this lane's channel (9 taps).
    // sign(v>=0)=+1 -> bit = (v < 0).
    // ---------------------------------------------------------------
    const int c     = tid & (CG_ - 1);         // 0..31
    const int wl    = tid >> 5;                // 0..7 (w lane group)
    const int cglob = cg * CG_ + c;
    unsigned kb[9];
    #pragma unroll
    for (int t = 0; t < 9; ++t)
        kb[t] = (k[t * C_ + cglob] < 0.0f) ? 1u : 0u;

    // ---------------------------------------------------------------
    // Compute 112x32 outputs: 14 w-positions per lane.
    // Out-of-range taps contribute 0 (zero padding of quantized input).
    // ---------------------------------------------------------------
    for (int i = 0; i < 14; ++i) {
        const int w = wl + i * 8;              // 0..111
        int acc = 0;
        #pragma unroll
        for (int r = 0; r < 3; ++r) {
            const int hh = h - 1 + r;
            if (hh < 0 || hh >= H_) continue;  // uniform per block
            #pragma unroll
            for (int tw = 0; tw < 3; ++tw) {
                const int ww = w - 1 + tw;
                if (ww < 0 || ww >= W_) continue;
                const unsigned xb =
                    (smem[(r * W_ + ww) * CG_ + c] < 0.0f) ? 1u : 0u;
                acc += (xb ^ kb[r * 3 + tw]) ? -1 : 1;
            }
        }
        out[((size_t)(n * H_ + h) * W_ + w) * C_ + cglob] = (float)acc;
    }
}

extern "C" void kernel_launch(void* const* d_in, const int* in_sizes, int n_in,
                              void* d_out, int out_size, void* d_ws, size_t ws_size,
                              hipStream_t stream) {
    const float* x = (const float*)d_in[0];   // (16,112,112,256) f32
    const float* k = (const float*)d_in[1];   // (3,3,256,1) f32
    float* out     = (float*)d_out;           // (16,112,112,256) f32
    (void)in_sizes; (void)n_in; (void)out_size; (void)d_ws; (void)ws_size;

    const int grid = N_ * H_ * NCG;           // 16 * 112 * 8 = 14336 workgroups
    bsign_dwconv3x3_kernel<<<grid, 256, 0, stream>>>(x, k, out);
}